// Embedding_28664611734255
// MI455X (gfx1250) — compile-verified
//
#include <hip/hip_runtime.h>
#include <stdint.h>

// out[t, :] = table[ids[t], :]   (t = 0 .. BATCH*SEQ-1, EMBED = VOCAB = 256, f32)
//
// Pure-bandwidth gather (roofline: ~256 MB NT stores at 23.3 TB/s ~= 11 us).
//   1. Stage the whole 256 KB table + this block's 2 KB of ids into LDS with
//      CDNA5 async global->LDS copies (ASYNCcnt DMA path, no VGPR staging).
//      The async op's 24-bit inst offset is added to BOTH the LDS and global
//      addresses, so 4 copies share one address-register pair.
//   2. Gather rows with conflict-free ds_load_b128 (32 lanes read 32
//      consecutive 16 B chunks of one row).
//   3. Stream results out with non-temporal b128 stores (write-once data).

typedef float v4f __attribute__((ext_vector_type(4)));

#define EMBED        256
#define ROW_BYTES    (EMBED * 4)            /* 1024 B per embedding row      */
#define TABLE_BYTES  (256 * ROW_BYTES)      /* 256 KB table, fits 320 KB LDS */
#define TPB          512                    /* tokens per block              */
#define IDS_BYTES    (TPB * 4)              /* 2 KB id stage                 */
#define LDS_BYTES    (TABLE_BYTES + IDS_BYTES)
#define THREADS      256

__global__ __launch_bounds__(THREADS)
void embed_gather_kernel(const int* __restrict__ ids,
                         const float* __restrict__ table,
                         float* __restrict__ out,
                         int n_tokens)
{
    extern __shared__ float lds_table[];    // 256 KB table + 2 KB ids

    const int tid     = threadIdx.x;
    const int t_begin = (int)blockIdx.x * TPB;
    int t_end         = t_begin + TPB;
    if (t_end > n_tokens) t_end = n_tokens;
    const int cnt     = t_end - t_begin;    // ids handled by this block

    // ---------------------------------------------------------------
    // Async-stage table -> LDS. 4 x b128 per group share one base pair
    // via the inst offset (added to both LDS and global addresses).
    // 16 groups x 16 KB = 256 KB.
    // ---------------------------------------------------------------
    {
        const uint64_t tb  = (uint64_t)(uintptr_t)table;
        uint32_t lds_off   = (uint32_t)(uintptr_t)lds_table + (uint32_t)tid * 16u;
        uint32_t gl_off    = (uint32_t)tid * 16u;
        for (int i = 0; i < TABLE_BYTES / (THREADS * 16 * 4); ++i) {
            asm volatile("global_load_async_to_lds_b128 %0, %1, %2"
                         :: "v"(lds_off), "v"(gl_off), "s"(tb) : "memory");
            asm volatile("global_load_async_to_lds_b128 %0, %1, %2 offset:4096"
                         :: "v"(lds_off), "v"(gl_off), "s"(tb) : "memory");
            asm volatile("global_load_async_to_lds_b128 %0, %1, %2 offset:8192"
                         :: "v"(lds_off), "v"(gl_off), "s"(tb) : "memory");
            asm volatile("global_load_async_to_lds_b128 %0, %1, %2 offset:12288"
                         :: "v"(lds_off), "v"(gl_off), "s"(tb) : "memory");
            lds_off += THREADS * 16u * 4u;
            gl_off  += THREADS * 16u * 4u;
        }

        // Async-stage this block's ids (b32 flavor) behind the table.
        const uint64_t ib = (uint64_t)(uintptr_t)ids;
        uint32_t ids_lds  = (uint32_t)(uintptr_t)lds_table + TABLE_BYTES
                          + (uint32_t)tid * 4u;
        uint32_t ids_gl   = ((uint32_t)t_begin + (uint32_t)tid) * 4u;
        for (int j = tid; j < cnt; j += THREADS) {
            asm volatile("global_load_async_to_lds_b32 %0, %1, %2"
                         :: "v"(ids_lds), "v"(ids_gl), "s"(ib) : "memory");
            ids_lds += THREADS * 4u;
            ids_gl  += THREADS * 4u;
        }

#if defined(__has_builtin) && __has_builtin(__builtin_amdgcn_s_wait_asynccnt)
        __builtin_amdgcn_s_wait_asynccnt(0);
#else
        asm volatile("s_wait_asynccnt 0x0" ::: "memory");
#endif
    }
    __syncthreads();

    // ---------------------------------------------------------------
    // Gather. 64 threads per row (1 KB); 4 tokens per block-iteration.
    // Per wave32: ds_load_b32 id (LDS-resident), conflict-free
    // ds_load_b128 row chunk, 512 B contiguous NT b128 store.
    // ---------------------------------------------------------------
    const int sub   = tid >> 6;   // 0..3  : token within group of 4
    const int chunk = tid & 63;   // 0..63 : float4 within row

    const v4f* __restrict__ lds4    = (const v4f*)lds_table;
    const int* __restrict__ lds_ids = (const int*)(lds_table + TABLE_BYTES / 4);

    #pragma unroll 4
    for (int j = sub; j < cnt; j += 4) {
        const int row = lds_ids[j];                    // ds_load_b32, 0..255
        v4f v = lds4[row * (ROW_BYTES / 16) + chunk];  // ds_load_b128
        __builtin_nontemporal_store(
            v, (v4f*)(out + (size_t)(t_begin + j) * EMBED + (size_t)chunk * 4));
    }
}

extern "C" void kernel_launch(void* const* d_in, const int* in_sizes, int n_in,
                              void* d_out, int out_size, void* d_ws, size_t ws_size,
                              hipStream_t stream)
{
    (void)d_ws; (void)ws_size; (void)n_in; (void)out_size;

    const int*   ids   = (const int*)d_in[0];     // [BATCH*SEQ] int32
    const float* table = (const float*)d_in[1];   // [VOCAB, EMBED] f32
    float*       out   = (float*)d_out;           // [BATCH*SEQ, EMBED] f32
    const int n_tokens = in_sizes[0];

    // 258 KB dynamic LDS > default cap; raise the limit (idempotent, safe
    // under graph capture: not a stream operation).
    (void)hipFuncSetAttribute((const void*)embed_gather_kernel,
                              hipFuncAttributeMaxDynamicSharedMemorySize,
                              LDS_BYTES);

    const int grid = (n_tokens + TPB - 1) / TPB;  // 512 for BATCH*SEQ = 262144
    embed_gather_kernel<<<dim3(grid), dim3(THREADS), LDS_BYTES, stream>>>(
        ids, table, out, n_tokens);
}